// TopKModule_48026324304303
// MI455X (gfx1250) — compile-verified
//
#include <hip/hip_runtime.h>
#include <hip/hip_bf16.h>

#define N_PTS   8192
#define KSEL    32
#define LISTLEN 33          // k+1: keep self, drop it at merge time
#define EPS_F   1e-8f

typedef __attribute__((ext_vector_type(2))) float v2f;
typedef __attribute__((ext_vector_type(8))) float v8f;

__global__ void knn_zero_grad(float* __restrict__ g, int n) {
    int i = blockIdx.x * blockDim.x + threadIdx.x;
    if (i < n) g[i] = 0.0f;
}

// One block = 16 rows. 4 waves (128 threads). Each wave computes one 16x16
// distance tile per chunk via V_WMMA_F32_16X16X4_F32 (Gram matrix of the
// 3D positions, K padded to 4), converts to distances, stages in LDS.
// Selection: 8 threads per row keep exact private sorted top-33 lists in LDS.
// Final: 8-way merge per row -> sorted dists + analytic gradient.
__global__ __launch_bounds__(128)
void knn_topk_grad(const float* __restrict__ pos,
                   float* __restrict__ outDist,
                   float* __restrict__ grad) {
    __shared__ float rowX[16], rowY[16], rowZ[16], rowR2[16];
    __shared__ float chunkD[16 * 64];                 // 16 rows x 64-col chunk
    __shared__ float listD[16 * 8 * LISTLEN];         // per (row, subthread) lists
    __shared__ int   listI[16 * 8 * LISTLEN];
    __shared__ int   listCnt[16 * 8];

    const int t    = threadIdx.x;
    const int lane = t & 31;
    const int wave = t >> 5;
    const int r0   = blockIdx.x * 16;

    if (t < 16) {
        const float x = pos[(r0 + t) * 3 + 0];
        const float y = pos[(r0 + t) * 3 + 1];
        const float z = pos[(r0 + t) * 3 + 2];
        rowX[t] = x; rowY[t] = y; rowZ[t] = z;
        rowR2[t] = x * x + y * y + z * z;
    }
    listCnt[t] = 0;
    __syncthreads();

    // A fragment: 16x4 f32, lane (l&15)=row M, lane half selects K pair (0,1)/(2,3)
    const int m    = lane & 15;
    const int half = lane >> 4;
    v2f afrag;
    afrag.x = half ? rowZ[m] : rowX[m];
    afrag.y = half ? 0.0f    : rowY[m];

    const int selRow = t >> 3;          // 0..15
    const int selS   = t & 7;           // 0..7 (8 selector threads per row)
    const int lbase  = (selRow * 8 + selS) * LISTLEN;

    for (int ch = 0; ch < N_PTS / 64; ++ch) {
        const int colBase = ch * 64;

        // ---- distance tile via WMMA (all 128 lanes active: EXEC all-ones) ----
        {
            const int   c   = colBase + wave * 16 + m;
            const float px  = pos[c * 3 + 0];
            const float py  = pos[c * 3 + 1];
            const float pz  = pos[c * 3 + 2];
            const float r2c = px * px + py * py + pz * pz;
            v2f bfrag;                   // B: 4x16, column (l&15)=N, half selects K pair
            bfrag.x = half ? pz : px;
            bfrag.y = half ? 0.0f : py;
            v8f acc = {};
            acc = __builtin_amdgcn_wmma_f32_16x16x4_f32(
                      false, afrag, false, bfrag, (short)0, acc, false, false);
#pragma unroll
            for (int r = 0; r < 8; ++r) {
                const int mr = r + 8 * half;          // C layout: M = r + 8*laneHalf
                float d2 = rowR2[mr] + r2c - 2.0f * acc[r] + EPS_F;
                d2 = fmaxf(d2, EPS_F);                // self-pair hits exactly eps
                chunkD[mr * 64 + wave * 16 + m] = sqrtf(d2);
            }
        }
        __syncthreads();

        // ---- exact per-thread top-33 insertion (8 cols per thread per chunk) ----
        {
            int cnt = listCnt[selRow * 8 + selS];
#pragma unroll
            for (int q = 0; q < 8; ++q) {
                const int   j  = selS + 8 * q;
                const float d  = chunkD[selRow * 64 + j];
                const int   gc = colBase + j;
                if (cnt < LISTLEN || d < listD[lbase + LISTLEN - 1]) {
                    int p = (cnt < LISTLEN) ? cnt : (LISTLEN - 1);
                    while (p > 0 && listD[lbase + p - 1] > d) {
                        listD[lbase + p] = listD[lbase + p - 1];
                        listI[lbase + p] = listI[lbase + p - 1];
                        --p;
                    }
                    listD[lbase + p] = d;
                    listI[lbase + p] = gc;
                    if (cnt < LISTLEN) ++cnt;
                }
            }
            listCnt[selRow * 8 + selS] = cnt;
        }
        __syncthreads();
    }

    // ---- 8-way merge per row; emit sorted dists + gradient ----
    if (t < 16) {
        const int row = t;
        const int gr  = r0 + row;
        int hp[8];
#pragma unroll
        for (int s = 0; s < 8; ++s) hp[s] = 0;
        const float rx = rowX[row], ry = rowY[row], rz = rowZ[row];
        float gx = 0.0f, gy = 0.0f, gz = 0.0f;

        for (int k = 0; k < LISTLEN; ++k) {
            float best = 3.4e38f;
            int   bs   = 0;
#pragma unroll
            for (int s = 0; s < 8; ++s) {
                const int bidx = (row * 8 + s) * LISTLEN;
                if (hp[s] < listCnt[row * 8 + s]) {
                    const float hd = listD[bidx + hp[s]];
                    if (hd < best) { best = hd; bs = s; }
                }
            }
            const int bidx = (row * 8 + bs) * LISTLEN;
            const int c    = listI[bidx + hp[bs]];
            hp[bs]++;
            if (k == 0) continue;                      // drop self column
            const int jj = k - 1;
            outDist[gr * KSEL + jj] = best;

            const float dx   = pos[c * 3 + 0] - rx;
            const float dy   = pos[c * 3 + 1] - ry;
            const float dz   = pos[c * 3 + 2] - rz;
            const float invd = 1.0f / best;
            gx -= dx * invd; gy -= dy * invd; gz -= dz * invd;   // d d/d pos_i
            atomicAdd(&grad[c * 3 + 0], dx * invd);              // d d/d pos_j
            atomicAdd(&grad[c * 3 + 1], dy * invd);
            atomicAdd(&grad[c * 3 + 2], dz * invd);
        }
        atomicAdd(&grad[gr * 3 + 0], gx);
        atomicAdd(&grad[gr * 3 + 1], gy);
        atomicAdd(&grad[gr * 3 + 2], gz);
    }
}

extern "C" void kernel_launch(void* const* d_in, const int* in_sizes, int n_in,
                              void* d_out, int out_size, void* d_ws, size_t ws_size,
                              hipStream_t stream) {
    (void)in_sizes; (void)n_in; (void)d_ws; (void)ws_size; (void)out_size;
    const float* positions = (const float*)d_in[0];          // [8192,3] f32
    float* out  = (float*)d_out;
    float* dist = out;                                       // [1, 8192*32]
    float* grad = out + (size_t)N_PTS * KSEL;                // [8192, 3]

    knn_zero_grad<<<(N_PTS * 3 + 255) / 256, 256, 0, stream>>>(grad, N_PTS * 3);
    knn_topk_grad<<<N_PTS / 16, 128, 0, stream>>>(positions, dist, grad);
}